// ObjectRankingLoss_71382356459605
// MI455X (gfx1250) — compile-verified
//
#include <hip/hip_runtime.h>
#include <stdint.h>

// Problem constants (from reference)
#define B_   4
#define H_   256
#define W_   832
#define NPX  (H_ * W_)        // 212992 pixels per batch
#define NI   15               // instance ids 1..15
#define WSZ  128              // pixels per window
#define NW   (NPX / WSZ)      // 1664 windows per batch
#define SIG  0.15f

// Workspace layout (bytes); total ~2.1 MB
#define ACC_OFF   0                                   // 2 doubles: [term_sum, w_sum]
#define MDM_OFF   256                                 // B*NW*4  u32 (dmask ballots)
#define MSAME_OFF (MDM_OFF + B_ * NW * 4 * 4)         // B*NI*NW*4 u32 (seg==id ballots)
#define CC_OFF    (MSAME_OFF + B_ * NI * NW * 4 * 4)  // B*NI*NW int32 (coarse incl. prefix)

#if __has_builtin(__builtin_amdgcn_global_load_async_to_lds_b128) && \
    __has_builtin(__builtin_amdgcn_s_wait_asynccnt)
#define USE_ASYNC_LDS 1
typedef int v4i_ __attribute__((vector_size(16)));             // matches builtin param
typedef __attribute__((address_space(1))) v4i_* as1_v4i;
typedef __attribute__((address_space(3))) v4i_* as3_v4i;
#else
#define USE_ASYNC_LDS 0
#endif

__global__ void k_init(double* acc) {
  if (threadIdx.x < 2) acc[threadIdx.x] = 0.0;
}

// Kernel A: build 128-bit window bitmasks via wave32 ballots.
// grid = (NW/8, B), block = 256 (8 waves, one window per wave, 4 px per lane)
__global__ __launch_bounds__(256) void k_masks(const float* __restrict__ gt,
                                               const int* __restrict__ seg,
                                               uint32_t* __restrict__ mdm,
                                               uint32_t* __restrict__ msame) {
  const int tid  = threadIdx.x;
  const int lane = tid & 31;
  const int wv   = tid >> 5;
  const int b    = blockIdx.y;
  const int j    = blockIdx.x * 8 + wv;        // window in [0, NW)
  const int gbase = b * NPX + j * WSZ;

  uint32_t dm[4];
  uint32_t mine[4] = {0u, 0u, 0u, 0u};
#pragma unroll
  for (int k = 0; k < 4; ++k) {
    const int   p = gbase + k * 32 + lane;
    const float d = gt[p];
    const int   s = seg[p];
    dm[k] = __builtin_amdgcn_ballot_w32(d > 0.0f);
#pragma unroll
    for (int id = 1; id <= NI; ++id) {
      const uint32_t bm = __builtin_amdgcn_ballot_w32(s == id);
      if (lane == id - 1) mine[k] = bm;        // lane id-1 keeps instance id's masks
    }
  }
  if (lane < NI) {
    uint4 v = make_uint4(mine[0], mine[1], mine[2], mine[3]);
    *(uint4*)&msame[(((size_t)b * NI + lane) * NW + j) * 4] = v;
  } else if (lane == NI) {
    uint4 v = make_uint4(dm[0], dm[1], dm[2], dm[3]);
    *(uint4*)&mdm[((size_t)b * NW + j) * 4] = v;
  }
}

// Kernel B: per (b,i) inclusive prefix sum of per-window valid counts.
// grid = B*NI, block = 256; 7 chunks of 256 cover 1664 windows.
__global__ __launch_bounds__(256) void k_coarse_scan(const uint32_t* __restrict__ mdm,
                                                     const uint32_t* __restrict__ msame,
                                                     int* __restrict__ cc) {
  const int tid = threadIdx.x, lane = tid & 31, wv = tid >> 5;
  const int bi  = blockIdx.x;                  // b*NI + i
  const int b   = bi / NI;
  __shared__ int wsum[8];
  int carry = 0;
  for (int c = 0; c < NW; c += 256) {
    const int j = c + tid;
    int v = 0;
    if (j < NW) {
      const uint4 md = *(const uint4*)&mdm[((size_t)b * NW + j) * 4];
      const uint4 ms = *(const uint4*)&msame[((size_t)bi * NW + j) * 4];
      v = __popc(md.x & ~ms.x) + __popc(md.y & ~ms.y) +
          __popc(md.z & ~ms.z) + __popc(md.w & ~ms.w);
    }
    int sc = v;                                // wave32 inclusive scan
#pragma unroll
    for (int d = 1; d < 32; d <<= 1) {
      const int t = __shfl_up(sc, d, 32);
      if (lane >= d) sc += t;
    }
    if (lane == 31) wsum[wv] = sc;
    __syncthreads();
    int off = carry;
    for (int q = 0; q < wv; ++q) off += wsum[q];
    if (j < NW) cc[(size_t)bi * NW + j] = sc + off;
    int tot = 0;
    for (int q = 0; q < 8; ++q) tot += wsum[q];
    carry += tot;
    __syncthreads();
  }
}

// Kernel C: main loss. grid = (NPX/256, B*NI), block = 256.
// Coarse table staged to LDS with CDNA5 async global->LDS B128 copies,
// then LDS binary search + popcount bit-select within a 128-bit window.
__global__ __launch_bounds__(256) void k_loss(const float* __restrict__ pred,
                                              const float* __restrict__ gt,
                                              const int* __restrict__ seg,
                                              const float* __restrict__ ru,
                                              const uint32_t* __restrict__ mdm,
                                              const uint32_t* __restrict__ msame,
                                              const int* __restrict__ cc,
                                              double* __restrict__ acc) {
  __shared__ int   sCum[NW];
  __shared__ float sT[8], sW[8];
  const int tid = threadIdx.x, lane = tid & 31, wv = tid >> 5;
  const int bi   = blockIdx.y;
  const int b    = bi / NI;
  const int inst = (bi % NI) + 1;

#if USE_ASYNC_LDS
  {
    // 1664 ints = 416 x B128 async transfers; no VGPR staging, ASYNCcnt tracked.
    const int* gsrc = &cc[(size_t)bi * NW];
    for (int t = tid; t < NW / 4; t += 256) {
      v4i_* gp = (v4i_*)(gsrc + 4 * t);        // drop const, re-type to 16B vector
      v4i_* lp = (v4i_*)&sCum[4 * t];
      __builtin_amdgcn_global_load_async_to_lds_b128(
          (as1_v4i)gp, (as3_v4i)lp, 0, 0);
    }
    __builtin_amdgcn_s_wait_asynccnt(0);   // my wave's async copies done
    __syncthreads();                        // all waves' copies visible
  }
#else
  for (int t = tid; t < NW; t += 256) sCum[t] = cc[(size_t)bi * NW + t];
  __syncthreads();
#endif
  const int cnt = sCum[NW - 1];

  const int   p     = blockIdx.x * 256 + tid;  // < NPX exactly
  const float za_gt = gt[b * NPX + p];
  const float za_pr = pred[b * NPX + p];
  const int   s     = seg[b * NPX + p];
  const float u     = ru[(size_t)bi * NPX + p];

  float term = 0.0f, wgt = 0.0f;
  const bool inval = (s == inst) && (za_gt > 0.0f);
  if (cnt > 0 && inval) {                      // weight can only be nonzero here
    int rank = (int)(u * (float)cnt);          // floor (u >= 0)
    rank = rank < 0 ? 0 : rank;
    rank = rank > cnt - 1 ? cnt - 1 : rank;
    // smallest j with sCum[j] > rank  (LDS binary search, 11 steps)
    int lo = 0, hi = NW;
    while (lo < hi) {
      const int mid = (lo + hi) >> 1;
      if (sCum[mid] > rank) hi = mid; else lo = mid + 1;
    }
    const int j    = lo;
    const int base = j ? sCum[j - 1] : 0;
    int need = rank - base;                    // 0-indexed valid within window
    const uint4 md = *(const uint4*)&mdm[((size_t)b * NW + j) * 4];
    const uint4 ms = *(const uint4*)&msame[((size_t)bi * NW + j) * 4];
    const uint32_t w0 = md.x & ~ms.x, w1 = md.y & ~ms.y;
    const uint32_t w2 = md.z & ~ms.z, w3 = md.w & ~ms.w;
    int k; uint32_t w;
    const int c0 = __popc(w0), c1 = __popc(w1), c2 = __popc(w2);
    if (need < c0)                { k = 0; w = w0; }
    else { need -= c0;
      if (need < c1)              { k = 1; w = w1; }
      else { need -= c1;
        if (need < c2)            { k = 2; w = w2; }
        else { need -= c2;          k = 3; w = w3; } } }
    // select need-th set bit of w (5 popcount steps)
    int pos = 0;
#pragma unroll
    for (int sh = 16; sh >= 1; sh >>= 1) {
      const uint32_t m  = ((1u << sh) - 1u) << pos;
      const int      cb = __popc(w & m);
      if (need >= cb) { need -= cb; pos += sh; }
    }
    int partner = j * WSZ + k * 32 + pos;
    if (partner > NPX - 1) partner = NPX - 1;
    const float zb_gt = gt[b * NPX + partner];
    const float zb_pr = pred[b * NPX + partner];
    const float tgt = (za_gt >= (1.0f + SIG) * zb_gt) ? 1.0f
                     : ((zb_gt > (1.0f + SIG) * za_gt) ? -1.0f : 0.0f);
    if (tgt != 0.0f) {
      wgt = 1.0f;
      const float x = -tgt * (za_pr - zb_pr);
      term = fmaxf(x, 0.0f) + log1pf(expf(-fabsf(x)));   // stable softplus
    }
  }
  // wave32 + LDS reduction, one f64 atomic pair per block
#pragma unroll
  for (int d = 16; d >= 1; d >>= 1) {
    term += __shfl_down(term, d, 32);
    wgt  += __shfl_down(wgt,  d, 32);
  }
  if (lane == 0) { sT[wv] = term; sW[wv] = wgt; }
  __syncthreads();
  if (tid == 0) {
    float t = 0.0f, wsum = 0.0f;
    for (int q = 0; q < 8; ++q) { t += sT[q]; wsum += sW[q]; }
    if (wsum > 0.0f || t != 0.0f) {
      atomicAdd(&acc[0], (double)t);
      atomicAdd(&acc[1], (double)wsum);
    }
  }
}

__global__ void k_fin(const double* __restrict__ acc, float* __restrict__ out) {
  if (threadIdx.x == 0) {
    const double d = acc[1];
    out[0] = (d > 0.0) ? (float)(acc[0] / d) : 0.0f;
  }
}

extern "C" void kernel_launch(void* const* d_in, const int* in_sizes, int n_in,
                              void* d_out, int out_size, void* d_ws, size_t ws_size,
                              hipStream_t stream) {
  (void)in_sizes; (void)n_in; (void)out_size; (void)ws_size;
  const float* pred = (const float*)d_in[0];   // pred_depth  (B,1,H,W) f32
  const float* gtd  = (const float*)d_in[1];   // gt_depth    (B,1,H,W) f32
  const int*   seg  = (const int*)d_in[2];     // seg_masks   (B,1,H,W) i32
  const float* ru   = (const float*)d_in[3];   // rand_u      (B,15,H*W) f32
  float* out = (float*)d_out;

  char* ws = (char*)d_ws;
  double*   acc   = (double*)(ws + ACC_OFF);
  uint32_t* mdm   = (uint32_t*)(ws + MDM_OFF);
  uint32_t* msame = (uint32_t*)(ws + MSAME_OFF);
  int*      cc    = (int*)(ws + CC_OFF);

  k_init<<<1, 32, 0, stream>>>(acc);
  k_masks<<<dim3(NW / 8, B_), 256, 0, stream>>>(gtd, seg, mdm, msame);
  k_coarse_scan<<<B_ * NI, 256, 0, stream>>>(mdm, msame, cc);
  k_loss<<<dim3(NPX / 256, B_ * NI), 256, 0, stream>>>(pred, gtd, seg, ru,
                                                       mdm, msame, cc, acc);
  k_fin<<<1, 32, 0, stream>>>(acc, out);
}